// MambaMIL_2D_81862076662076
// MI455X (gfx1250) — compile-verified
//
#include <hip/hip_runtime.h>
#include <hip/hip_bf16.h>

// ---------------- model dims ----------------
#define D_MODEL   128
#define D_INNER   256
#define D_STATE   16
#define DT_RANK   8
#define D_CONV    4
#define N_LAYERS  2
#define GRID_     128
#define PATCH     512
#define N_TOK     12000
#define DIM_IN    1024
#define N_CLASSES 2
#define EPS_F     1e-5f
#define SEQ_L     (GRID_*GRID_)        /* 16384 */
#define CHUNK     128
#define NCHUNK    (SEQ_L / CHUNK)      /* 128 */

typedef __attribute__((ext_vector_type(2))) float v2f;
typedef __attribute__((ext_vector_type(8))) float v8f;

// ---------------- small math helpers ----------------
__device__ __forceinline__ float silu_f(float x) { return x / (1.0f + expf(-x)); }
__device__ __forceinline__ float gelu_f(float x) { return 0.5f * x * (1.0f + erff(x * 0.70710678118654752f)); }
__device__ __forceinline__ float softplus_f(float x) { return (x > 20.0f) ? x : log1pf(expf(x)); }

// ---------------- fp32 WMMA helpers ----------------
// A frag (16x4/step): lanes 0-15 -> M=lane, K {k0,k0+1}; lanes 16-31 -> M=lane-16, K {k0+2,k0+3}
// B frag (4x16/step): lanes 0-15 -> N=lane, rows {k0,k0+1}; lanes 16-31 -> N=lane-16, rows {k0+2,k0+3}
// D: VGPR rr: lanes 0-15 -> M=m0+rr, N=n0+lane; lanes 16-31 -> M=m0+8+rr, N=n0+lane-16

// 16x32 strip per wave: two accumulators share one A fragment. N must be >= n0+32; no guard.
__device__ __forceinline__ void wmma_strip32_f32(const float* __restrict__ A, int lda,
                                                 const float* __restrict__ B, int ldb,
                                                 int K, int m0, int n0, int lane,
                                                 v8f& acc0, v8f& acc1)
{
  const int r    = lane & 15;
  const int koff = (lane >> 4) << 1;
  const float* ap = A + (m0 + r) * lda + koff;
  for (int k0 = 0; k0 < K; k0 += 4) {
    v2f a;
    a.x = ap[k0];
    a.y = ap[k0 + 1];
    const float* brow0 = B + (k0 + koff) * ldb + n0 + r;
    const float* brow1 = brow0 + ldb;
    v2f b0, b1;
    b0.x = brow0[0];  b0.y = brow1[0];
    b1.x = brow0[16]; b1.y = brow1[16];
    acc0 = __builtin_amdgcn_wmma_f32_16x16x4_f32(false, a, false, b0, (short)0, acc0, false, false);
    acc1 = __builtin_amdgcn_wmma_f32_16x16x4_f32(false, a, false, b1, (short)0, acc1, false, false);
  }
}

// single 16x16 tile with branch-free ragged-N guard (clamped index + select, no EXEC churn)
__device__ __forceinline__ v8f wmma_tile_guarded_f32(const float* __restrict__ A, int lda,
                                                     const float* __restrict__ B, int ldb,
                                                     int K, int m0, int n0, int lane, int nvalid)
{
  v8f acc = {};
  const int r    = lane & 15;
  const int koff = (lane >> 4) << 1;
  const int nc   = n0 + r;
  const bool ok  = (nc < nvalid);
  const int ncc  = ok ? nc : 0;
  const float msk = ok ? 1.0f : 0.0f;
  const float* ap = A + (m0 + r) * lda + koff;
  for (int k0 = 0; k0 < K; k0 += 4) {
    v2f a, b;
    a.x = ap[k0];
    a.y = ap[k0 + 1];
    b.x = B[(k0 + koff) * ldb + ncc] * msk;
    b.y = B[(k0 + koff + 1) * ldb + ncc] * msk;
    acc = __builtin_amdgcn_wmma_f32_16x16x4_f32(false, a, false, b, (short)0, acc, false, false);
  }
  return acc;
}

// ---------------- kernels ----------------

// h = gelu(feats @ W_fc1 + b_fc1)   [12000 x 1024] x [1024 x 128]; 750 x 4 strips
__global__ void k_fc1_gelu(const float* __restrict__ A, const float* __restrict__ B,
                           const float* __restrict__ bias, float* __restrict__ C)
{
  int wave = blockIdx.x * (blockDim.x >> 5) + (threadIdx.x >> 5);
  int lane = threadIdx.x & 31;
  const int Nstrips = D_MODEL / 32;
  int mt = wave / Nstrips, nt = wave % Nstrips;
  v8f acc0 = {}, acc1 = {};
  wmma_strip32_f32(A, DIM_IN, B, D_MODEL, DIM_IN, mt * 16, nt * 32, lane, acc0, acc1);
  int col = nt * 32 + (lane & 15);
  int rb  = mt * 16 + ((lane >> 4) << 3);
  float bv0 = bias[col], bv1 = bias[col + 16];
  #pragma unroll
  for (int rr = 0; rr < 8; ++rr) {
    C[(rb + rr) * D_MODEL + col]      = gelu_f(acc0[rr] + bv0);
    C[(rb + rr) * D_MODEL + col + 16] = gelu_f(acc1[rr] + bv1);
  }
}

// generic full-width GEMM (N multiple of 32): C = A @ B
__global__ void k_gemm_plain(const float* __restrict__ A, int lda,
                             const float* __restrict__ B, int ldb,
                             float* __restrict__ C, int ldc,
                             int K, int Nstrips)
{
  int wave = blockIdx.x * (blockDim.x >> 5) + (threadIdx.x >> 5);
  int lane = threadIdx.x & 31;
  int mt = wave / Nstrips, nt = wave % Nstrips;
  v8f acc0 = {}, acc1 = {};
  wmma_strip32_f32(A, lda, B, ldb, K, mt * 16, nt * 32, lane, acc0, acc1);
  int col = nt * 32 + (lane & 15);
  int rb  = mt * 16 + ((lane >> 4) << 3);
  #pragma unroll
  for (int rr = 0; rr < 8; ++rr) {
    C[(rb + rr) * ldc + col]      = acc0[rr];
    C[(rb + rr) * ldc + col + 16] = acc1[rr];
  }
}

// x_proj: proj = xc @ x_proj_w, N=40 (ragged, guarded), 1024 x 3 tiles
__global__ void k_xproj(const float* __restrict__ A, const float* __restrict__ B,
                        float* __restrict__ C)
{
  int wave = blockIdx.x * (blockDim.x >> 5) + (threadIdx.x >> 5);
  int lane = threadIdx.x & 31;
  const int Ntiles = 3;                  /* 48 cols padded from 40 */
  const int NV = DT_RANK + 2 * D_STATE;  /* 40 */
  int mt = wave / Ntiles, nt = wave % Ntiles;
  v8f acc = wmma_tile_guarded_f32(A, D_INNER, B, NV, D_INNER, mt * 16, nt * 16, lane, NV);
  int col = nt * 16 + (lane & 15);
  int rb  = mt * 16 + ((lane >> 4) << 3);
  if (col < NV) {
    #pragma unroll
    for (int rr = 0; rr < 8; ++rr)
      C[(rb + rr) * NV + col] = acc[rr];
  }
}

// dt = softplus(proj[:, :8] @ dt_proj_w + dt_proj_b)   K=8, N=256 -> 1024 x 8 strips
__global__ void k_dtproj(const float* __restrict__ proj, const float* __restrict__ W,
                         const float* __restrict__ bias, float* __restrict__ dt)
{
  int wave = blockIdx.x * (blockDim.x >> 5) + (threadIdx.x >> 5);
  int lane = threadIdx.x & 31;
  const int Nstrips = D_INNER / 32;
  int mt = wave / Nstrips, nt = wave % Nstrips;
  v8f acc0 = {}, acc1 = {};
  wmma_strip32_f32(proj, DT_RANK + 2 * D_STATE, W, D_INNER, DT_RANK,
                   mt * 16, nt * 32, lane, acc0, acc1);
  int col = nt * 32 + (lane & 15);
  int rb  = mt * 16 + ((lane >> 4) << 3);
  float bv0 = bias[col], bv1 = bias[col + 16];
  #pragma unroll
  for (int rr = 0; rr < 8; ++rr) {
    dt[(rb + rr) * D_INNER + col]      = softplus_f(acc0[rr] + bv0);
    dt[(rb + rr) * D_INNER + col + 16] = softplus_f(acc1[rr] + bv1);
  }
}

// seq += y @ out_proj_w   (residual add), N=128 -> 1024 x 4 strips
__global__ void k_outproj_res(const float* __restrict__ Y, const float* __restrict__ W,
                              float* __restrict__ seq)
{
  int wave = blockIdx.x * (blockDim.x >> 5) + (threadIdx.x >> 5);
  int lane = threadIdx.x & 31;
  const int Nstrips = D_MODEL / 32;
  int mt = wave / Nstrips, nt = wave % Nstrips;
  v8f acc0 = {}, acc1 = {};
  wmma_strip32_f32(Y, D_INNER, W, D_MODEL, D_INNER, mt * 16, nt * 32, lane, acc0, acc1);
  int col = nt * 32 + (lane & 15);
  int rb  = mt * 16 + ((lane >> 4) << 3);
  #pragma unroll
  for (int rr = 0; rr < 8; ++rr) {
    seq[(rb + rr) * D_MODEL + col]      += acc0[rr];
    seq[(rb + rr) * D_MODEL + col + 16] += acc1[rr];
  }
}

// t = tanh(hn @ attn_w1 + attn_b1), N=128 -> 1024 x 4 strips
__global__ void k_attn1(const float* __restrict__ HN, const float* __restrict__ W,
                        const float* __restrict__ bias, float* __restrict__ T)
{
  int wave = blockIdx.x * (blockDim.x >> 5) + (threadIdx.x >> 5);
  int lane = threadIdx.x & 31;
  const int Nstrips = D_MODEL / 32;
  int mt = wave / Nstrips, nt = wave % Nstrips;
  v8f acc0 = {}, acc1 = {};
  wmma_strip32_f32(HN, D_MODEL, W, D_MODEL, D_MODEL, mt * 16, nt * 32, lane, acc0, acc1);
  int col = nt * 32 + (lane & 15);
  int rb  = mt * 16 + ((lane >> 4) << 3);
  float bv0 = bias[col], bv1 = bias[col + 16];
  #pragma unroll
  for (int rr = 0; rr < 8; ++rr) {
    T[(rb + rr) * D_MODEL + col]      = tanhf(acc0[rr] + bv0);
    T[(rb + rr) * D_MODEL + col + 16] = tanhf(acc1[rr] + bv1);
  }
}

// seq[g,:] = pad_token
__global__ void k_init_seq(const float* __restrict__ pad, float* __restrict__ seq)
{
  seq[blockIdx.x * D_MODEL + threadIdx.x] = pad[threadIdx.x];
}

// seq[gidx(n),:] = h[n,:]
__global__ void k_scatter(const float* __restrict__ h, const int* __restrict__ coords,
                          float* __restrict__ seq)
{
  int n = blockIdx.x;
  int g = (coords[2 * n] / PATCH) * GRID_ + (coords[2 * n + 1] / PATCH);
  seq[g * D_MODEL + threadIdx.x] = h[n * D_MODEL + threadIdx.x];
}

// RMSNorm over last dim (128), one row per block
__global__ void k_rmsnorm(const float* __restrict__ x, const float* __restrict__ w,
                          float* __restrict__ o)
{
  __shared__ float red[D_MODEL];
  int row = blockIdx.x, t = threadIdx.x;
  float v = x[row * D_MODEL + t];
  red[t] = v * v;
  __syncthreads();
  for (int s = D_MODEL / 2; s > 0; s >>= 1) { if (t < s) red[t] += red[t + s]; __syncthreads(); }
  float scale = rsqrtf(red[0] / (float)D_MODEL + EPS_F);
  o[row * D_MODEL + t] = v * scale * w[t];
}

// LayerNorm over last dim (128)
__global__ void k_layernorm(const float* __restrict__ x, const float* __restrict__ w,
                            const float* __restrict__ b, float* __restrict__ o)
{
  __shared__ float red[D_MODEL];
  int row = blockIdx.x, t = threadIdx.x;
  float v = x[row * D_MODEL + t];
  red[t] = v;
  __syncthreads();
  for (int s = D_MODEL / 2; s > 0; s >>= 1) { if (t < s) red[t] += red[t + s]; __syncthreads(); }
  float mu = red[0] / (float)D_MODEL;
  __syncthreads();
  float dv = v - mu;
  red[t] = dv * dv;
  __syncthreads();
  for (int s = D_MODEL / 2; s > 0; s >>= 1) { if (t < s) red[t] += red[t + s]; __syncthreads(); }
  float var = red[0] / (float)D_MODEL;
  o[row * D_MODEL + t] = dv * rsqrtf(var + EPS_F) * w[t] + b[t];
}

// causal depthwise conv (D_CONV=4) + SiLU; xm = xz[:, :256]
__global__ void k_conv_silu(const float* __restrict__ xz, const float* __restrict__ cw,
                            const float* __restrict__ cb, float* __restrict__ xc)
{
  int idx = blockIdx.x * blockDim.x + threadIdx.x;
  if (idx >= SEQ_L * D_INNER) return;
  int l = idx / D_INNER, d = idx % D_INNER;
  float acc = cb[d];
  #pragma unroll
  for (int k = 0; k < D_CONV; ++k) {
    int lp = l + k - (D_CONV - 1);
    if (lp >= 0) acc += xz[lp * (2 * D_INNER) + d] * cw[d * D_CONV + k];
  }
  xc[idx] = silu_f(acc);
}

// scan pass 1: per (chunk, d) run recurrence from zero state; record prodA = exp(A*sum(dt)), hend
__global__ void k_scan1(const float* __restrict__ dt, const float* __restrict__ xc,
                        const float* __restrict__ proj, const float* __restrict__ A_log,
                        float* __restrict__ prodA, float* __restrict__ hend)
{
  int c = blockIdx.x, d = threadIdx.x;
  float Aln[D_STATE], hst[D_STATE];
  #pragma unroll
  for (int s = 0; s < D_STATE; ++s) { Aln[s] = -expf(A_log[d * D_STATE + s]); hst[s] = 0.0f; }
  float dtsum = 0.0f;
  int l0 = c * CHUNK;
  for (int l = l0; l < l0 + CHUNK; ++l) {
    float e  = dt[l * D_INNER + d];
    float dx = e * xc[l * D_INNER + d];
    dtsum += e;
    const float* pr = proj + l * (DT_RANK + 2 * D_STATE);
    #pragma unroll
    for (int s = 0; s < D_STATE; ++s)
      hst[s] = expf(e * Aln[s]) * hst[s] + dx * pr[DT_RANK + s];
  }
  int base = (c * D_INNER + d) * D_STATE;
  #pragma unroll
  for (int s = 0; s < D_STATE; ++s) { prodA[base + s] = expf(Aln[s] * dtsum); hend[base + s] = hst[s]; }
}

// scan pass 2: serial combine of 128 chunk summaries per (d,s) -> carry-in state h0
__global__ void k_scan2(const float* __restrict__ prodA, const float* __restrict__ hend,
                        float* __restrict__ h0)
{
  int ds = blockIdx.x * blockDim.x + threadIdx.x;
  if (ds >= D_INNER * D_STATE) return;
  float carry = 0.0f;
  for (int c = 0; c < NCHUNK; ++c) {
    int idx = c * D_INNER * D_STATE + ds;
    h0[idx] = carry;
    carry = prodA[idx] * carry + hend[idx];
  }
}

// scan pass 3: rerun chunk with carry-in; fuse C-contraction, D skip, SiLU(z) gate
__global__ void k_scan3(const float* __restrict__ dt, const float* __restrict__ xc,
                        const float* __restrict__ proj, const float* __restrict__ A_log,
                        const float* __restrict__ Dp, const float* __restrict__ h0,
                        const float* __restrict__ xz, float* __restrict__ y)
{
  int c = blockIdx.x, d = threadIdx.x;
  float Aln[D_STATE], hst[D_STATE];
  int base = (c * D_INNER + d) * D_STATE;
  #pragma unroll
  for (int s = 0; s < D_STATE; ++s) { Aln[s] = -expf(A_log[d * D_STATE + s]); hst[s] = h0[base + s]; }
  float Dd = Dp[d];
  int l0 = c * CHUNK;
  for (int l = l0; l < l0 + CHUNK; ++l) {
    float e  = dt[l * D_INNER + d];
    float xv = xc[l * D_INNER + d];
    float dx = e * xv;
    const float* pr = proj + l * (DT_RANK + 2 * D_STATE);
    float acc = 0.0f;
    #pragma unroll
    for (int s = 0; s < D_STATE; ++s) {
      hst[s] = expf(e * Aln[s]) * hst[s] + dx * pr[DT_RANK + s];
      acc += hst[s] * pr[DT_RANK + D_STATE + s];
    }
    acc += Dd * xv;
    float z = xz[l * (2 * D_INNER) + D_INNER + d];
    y[l * D_INNER + d] = acc * silu_f(z);
  }
}

// a[l] = t[l,:] @ attn_w2 + attn_b2
__global__ void k_attn2(const float* __restrict__ T, const float* __restrict__ w2,
                        const float* __restrict__ b2, float* __restrict__ a)
{
  int l = blockIdx.x * blockDim.x + threadIdx.x;
  if (l >= SEQ_L) return;
  float acc = b2[0];
  for (int j = 0; j < D_MODEL; ++j) acc += T[l * D_MODEL + j] * w2[j];
  a[l] = acc;
}

// softmax over a, attention-pool hn, classifier head -> out[2]
__global__ void k_pool_clf(const float* __restrict__ a, const float* __restrict__ hn,
                           const float* __restrict__ clf_w, const float* __restrict__ clf_b,
                           float* __restrict__ out)
{
  __shared__ float red[D_MODEL];
  __shared__ float pooled[D_MODEL];
  int t = threadIdx.x;
  float mx = -1e30f;
  for (int l = t; l < SEQ_L; l += D_MODEL) mx = fmaxf(mx, a[l]);
  red[t] = mx;
  __syncthreads();
  for (int s = D_MODEL / 2; s > 0; s >>= 1) { if (t < s) red[t] = fmaxf(red[t], red[t + s]); __syncthreads(); }
  float gmx = red[0];
  __syncthreads();
  float sm = 0.0f;
  for (int l = t; l < SEQ_L; l += D_MODEL) sm += expf(a[l] - gmx);
  red[t] = sm;
  __syncthreads();
  for (int s = D_MODEL / 2; s > 0; s >>= 1) { if (t < s) red[t] += red[t + s]; __syncthreads(); }
  float Z = red[0];
  float pj = 0.0f;
  for (int l = 0; l < SEQ_L; ++l) pj += expf(a[l] - gmx) * hn[l * D_MODEL + t];
  pooled[t] = pj / Z;
  __syncthreads();
  if (t < N_CLASSES) {
    float acc = clf_b[t];
    for (int j = 0; j < D_MODEL; ++j) acc += pooled[j] * clf_w[j * N_CLASSES + t];
    out[t] = acc;
  }
}

// ---------------- host launcher ----------------
extern "C" void kernel_launch(void* const* d_in, const int* in_sizes, int n_in,
                              void* d_out, int out_size, void* d_ws, size_t ws_size,
                              hipStream_t stream)
{
  const float* feats      = (const float*)d_in[0];
  const int*   coords     = (const int*)  d_in[1];
  const float* W_fc1      = (const float*)d_in[2];
  const float* b_fc1      = (const float*)d_in[3];
  const float* pad_token  = (const float*)d_in[4];
  const float* rms_w      = (const float*)d_in[5];
  const float* in_proj_w  = (const float*)d_in[6];
  const float* conv_w     = (const float*)d_in[7];
  const float* conv_b     = (const float*)d_in[8];
  const float* x_proj_w   = (const float*)d_in[9];
  const float* dt_proj_w  = (const float*)d_in[10];
  const float* dt_proj_b  = (const float*)d_in[11];
  const float* A_log      = (const float*)d_in[12];
  const float* Dp         = (const float*)d_in[13];
  const float* out_proj_w = (const float*)d_in[14];
  const float* ln_w       = (const float*)d_in[15];
  const float* ln_b       = (const float*)d_in[16];
  const float* attn_w1    = (const float*)d_in[17];
  const float* attn_b1    = (const float*)d_in[18];
  const float* attn_w2    = (const float*)d_in[19];
  const float* attn_b2    = (const float*)d_in[20];
  const float* clf_w      = (const float*)d_in[21];
  const float* clf_b      = (const float*)d_in[22];
  float* out = (float*)d_out;

  // workspace carve-up (floats); total ~133 MB
  float* ws = (float*)d_ws;
  size_t off = 0;
  float* h      = ws + off; off += (size_t)N_TOK * D_MODEL;
  float* seq    = ws + off; off += (size_t)SEQ_L * D_MODEL;
  float* normed = ws + off; off += (size_t)SEQ_L * D_MODEL;
  float* xz     = ws + off; off += (size_t)SEQ_L * 2 * D_INNER;
  float* xc     = ws + off; off += (size_t)SEQ_L * D_INNER;
  float* proj   = ws + off; off += (size_t)SEQ_L * (DT_RANK + 2 * D_STATE);
  float* dt     = ws + off; off += (size_t)SEQ_L * D_INNER;
  float* yb     = ws + off; off += (size_t)SEQ_L * D_INNER;
  float* prodA  = ws + off; off += (size_t)NCHUNK * D_INNER * D_STATE;
  float* hend   = ws + off; off += (size_t)NCHUNK * D_INNER * D_STATE;
  float* h0     = ws + off; off += (size_t)NCHUNK * D_INNER * D_STATE;
  float* hn     = ws + off; off += (size_t)SEQ_L * D_MODEL;
  float* tb     = ws + off; off += (size_t)SEQ_L * D_MODEL;
  float* ab     = ws + off; off += (size_t)SEQ_L;
  (void)off; (void)ws_size; (void)in_sizes; (void)n_in; (void)out_size;

  // 1) fc1 + GELU: 750 x 4 = 3000 strip-waves, 4 waves/block
  k_fc1_gelu<<<750, 128, 0, stream>>>(feats, W_fc1, b_fc1, h);

  // 2) build grid sequence
  k_init_seq<<<SEQ_L, D_MODEL, 0, stream>>>(pad_token, seq);
  k_scatter<<<N_TOK, D_MODEL, 0, stream>>>(h, coords, seq);

  // 3) Mamba layers
  for (int i = 0; i < N_LAYERS; ++i) {
    k_rmsnorm<<<SEQ_L, D_MODEL, 0, stream>>>(seq, rms_w + i * D_MODEL, normed);
    // in_proj: 1024 x 16 strips
    k_gemm_plain<<<4096, 128, 0, stream>>>(normed, D_MODEL,
                                           in_proj_w + (size_t)i * D_MODEL * 2 * D_INNER, 2 * D_INNER,
                                           xz, 2 * D_INNER, D_MODEL, 16);
    k_conv_silu<<<(SEQ_L * D_INNER) / 256, 256, 0, stream>>>(xz, conv_w + (size_t)i * D_INNER * D_CONV,
                                                             conv_b + (size_t)i * D_INNER, xc);
    // x_proj: 1024 x 3 tiles (N=40 padded to 48, branch-free guard)
    k_xproj<<<768, 128, 0, stream>>>(xc, x_proj_w + (size_t)i * D_INNER * (DT_RANK + 2 * D_STATE), proj);
    // dt_proj: 1024 x 8 strips, K=8
    k_dtproj<<<2048, 128, 0, stream>>>(proj, dt_proj_w + (size_t)i * DT_RANK * D_INNER,
                                       dt_proj_b + (size_t)i * D_INNER, dt);
    // chunked selective scan
    k_scan1<<<NCHUNK, D_INNER, 0, stream>>>(dt, xc, proj, A_log + (size_t)i * D_INNER * D_STATE,
                                            prodA, hend);
    k_scan2<<<(D_INNER * D_STATE) / 256, 256, 0, stream>>>(prodA, hend, h0);
    k_scan3<<<NCHUNK, D_INNER, 0, stream>>>(dt, xc, proj, A_log + (size_t)i * D_INNER * D_STATE,
                                            Dp + (size_t)i * D_INNER, h0, xz, yb);
    // out_proj + residual: 1024 x 4 strips
    k_outproj_res<<<1024, 128, 0, stream>>>(yb, out_proj_w + (size_t)i * D_INNER * D_MODEL, seq);
  }

  // 4) head
  k_layernorm<<<SEQ_L, D_MODEL, 0, stream>>>(seq, ln_w, ln_b, hn);
  k_attn1<<<1024, 128, 0, stream>>>(hn, attn_w1, attn_b1, tb);
  k_attn2<<<SEQ_L / 256, 256, 0, stream>>>(tb, attn_w2, attn_b2, ab);
  k_pool_clf<<<1, D_MODEL, 0, stream>>>(ab, hn, clf_w, clf_b, out);
}